// RNNModel_38766374814114
// MI455X (gfx1250) — compile-verified
//
#include <hip/hip_runtime.h>

typedef __attribute__((ext_vector_type(2))) float v2f;
typedef __attribute__((ext_vector_type(8))) float v8f;

__device__ __forceinline__ float fast_tanh(float x) {
#if __has_builtin(__builtin_amdgcn_tanhf)
  return __builtin_amdgcn_tanhf(x);          // v_tanh_f32 (gfx1250 trans op)
#else
  // tanh(x) = 1 - 2/(exp(2x)+1) via v_exp_f32/v_rcp_f32; saturates to +-1.
  float e = __builtin_amdgcn_exp2f(x * 2.8853900817779268f); // 2*log2(e)
  return 1.0f - 2.0f * __builtin_amdgcn_rcpf(e + 1.0f);
#endif
}

__launch_bounds__(32)
__global__ void rnn_wmma_kernel(const float* __restrict__ x,
                                const float* __restrict__ W_ih,
                                const float* __restrict__ W_hh,
                                const float* __restrict__ b_ih,
                                const float* __restrict__ b_hh,
                                const float* __restrict__ fc_w,
                                const float* __restrict__ fc_b,
                                float* __restrict__ out,
                                int T) {
  const int lane = threadIdx.x & 31;
  const int base = blockIdx.x * 16;    // one wave per workgroup, 16 batch rows
  const int n    = lane & 15;          // batch column, duplicated across lane halves
  const int b    = base + n;

  // ---- uniform weights -> SGPRs ----
  float wih[8][3], bias[8];
  #pragma unroll
  for (int r = 0; r < 8; ++r) {
    wih[r][0] = W_ih[r * 3 + 0];
    wih[r][1] = W_ih[r * 3 + 1];
    wih[r][2] = W_ih[r * 3 + 2];
    bias[r]   = b_ih[r] + b_hh[r];
  }

  // ---- A operands: W_hh as 16x4 f32 A-matrix, K split 0..3 / 4..7 ----
  // A layout (16x4 f32): lanes 0-15: v0=K0, v1=K1 ; lanes 16-31: v0=K2, v1=K3 (M = lane%16).
  // Rows 8..15 duplicate rows 0..7 so both C/D lane-halves hold identical results:
  // the next-step B operand needs only 4 v_cndmask, no cross-lane permutes.
  const int half = lane >> 4;
  const int row  = lane & 7;
  v2f a1, a2;
  a1.x = W_hh[row * 8 + (half ? 2 : 0)];
  a1.y = W_hh[row * 8 + (half ? 3 : 1)];
  a2.x = W_hh[row * 8 + (half ? 6 : 4)];
  a2.y = W_hh[row * 8 + (half ? 7 : 5)];

  // recurrent state: th[r] = tanh hidden unit r of batch column n (same in both halves)
  float th[8];
  #pragma unroll
  for (int r = 0; r < 8; ++r) th[r] = 0.0f;

  const float* xb = x + (size_t)b * (size_t)T * 3u;

  // ---- 4-deep software pipeline on the x stream ----
  float px0[4], px1[4], px2[4];
  #pragma unroll
  for (int k = 0; k < 4; ++k) {
    px0[k] = xb[3 * k + 0];
    px1[k] = xb[3 * k + 1];
    px2[k] = xb[3 * k + 2];
  }

  for (int t0 = 0; t0 < T; t0 += 4) {
    #pragma unroll
    for (int k = 0; k < 4; ++k) {
      const int t = t0 + k;
      float cx0 = px0[k], cx1 = px1[k], cx2 = px2[k];

      int tn = t + 4;                       // uniform clamp: no OOB prefetch
      if (tn > T - 1) tn = T - 1;
      px0[k] = xb[3 * tn + 0];
      px1[k] = xb[3 * tn + 1];
      px2[k] = xb[3 * tn + 2];

      // C := x_t @ W_ih^T + (b_ih + b_hh), valid in both lane halves
      v8f c;
      #pragma unroll
      for (int r = 0; r < 8; ++r)
        c[r] = __builtin_fmaf(wih[r][0], cx0,
               __builtin_fmaf(wih[r][1], cx1,
               __builtin_fmaf(wih[r][2], cx2, bias[r])));

      // B operands (4x16 f32): lanes 0-15: v0=K0, v1=K1 ; lanes 16-31: v0=K2, v1=K3
      v2f b1, b2;
      b1.x = (half == 0) ? th[0] : th[2];
      b1.y = (half == 0) ? th[1] : th[3];
      b2.x = (half == 0) ? th[4] : th[6];
      b2.y = (half == 0) ? th[5] : th[7];

      // D = W_hh * h + C, K=8 as two chained K=4 f32 WMMAs
      c = __builtin_amdgcn_wmma_f32_16x16x4_f32(false, a1, false, b1, (short)0, c, false, false);
      c = __builtin_amdgcn_wmma_f32_16x16x4_f32(false, a2, false, b2, (short)0, c, false, false);

      #pragma unroll
      for (int r = 0; r < 8; ++r) th[r] = fast_tanh(c[r]);
    }
  }

  // ---- final FC: out[b][o] = sum_r fc_w[o][r]*h[r] + fc_b[o]  (O=2) ----
  if (lane < 16) {
    float o0 = fc_b[0], o1 = fc_b[1];
    #pragma unroll
    for (int r = 0; r < 8; ++r) {
      o0 = __builtin_fmaf(fc_w[r],     th[r], o0);
      o1 = __builtin_fmaf(fc_w[8 + r], th[r], o1);
    }
    out[2 * b + 0] = o0;
    out[2 * b + 1] = o1;
  }
}

extern "C" void kernel_launch(void* const* d_in, const int* in_sizes, int n_in,
                              void* d_out, int out_size, void* d_ws, size_t ws_size,
                              hipStream_t stream) {
  const float* x    = (const float*)d_in[0];
  const float* W_ih = (const float*)d_in[1];
  const float* W_hh = (const float*)d_in[2];
  const float* b_ih = (const float*)d_in[3];
  const float* b_hh = (const float*)d_in[4];
  const float* fc_w = (const float*)d_in[5];
  const float* fc_b = (const float*)d_in[6];
  float* out = (float*)d_out;

  const int T = 1024, I = 3;
  const int B = in_sizes[0] / (T * I);   // 8192
  const int waves = B / 16;              // 512 waves, 16 batch rows each

  // Single-wave workgroups: spread 512 independent latency-bound chains
  // across as many WGPs/SIMDs as the device has, instead of packing
  // 8 waves into one WGP.
  rnn_wmma_kernel<<<waves, 32, 0, stream>>>(x, W_ih, W_hh, b_ih, b_hh,
                                            fc_w, fc_b, out, T);
}